// GCN_74345883894238
// MI455X (gfx1250) — compile-verified
//
#include <hip/hip_runtime.h>

typedef float v2f __attribute__((ext_vector_type(2)));
typedef float v8f __attribute__((ext_vector_type(8)));

static constexpr int NN  = 100000;   // nodes
static constexpr int NE  = 1600000;  // edges
static constexpr int NF  = 128;      // feat / hidden (GEMM K is always 128)
static constexpr int NC  = 40;       // classes
static constexpr int NCP = 48;       // padded class dim for 16x16 tiling

// ---------------- graph preprocessing ----------------

__global__ void k_init(int* __restrict__ cnt, int* __restrict__ cur, int n) {
  int i = blockIdx.x * blockDim.x + threadIdx.x;
  if (i < n) { cnt[i] = 0; cur[i] = 0; }
}

__global__ void k_count(const int* __restrict__ dst, int* __restrict__ cnt, int e) {
  int i = blockIdx.x * blockDim.x + threadIdx.x;
  if (i < e) atomicAdd(&cnt[dst[i]], 1);
}

__global__ void k_dinv(const int* __restrict__ cnt, float* __restrict__ dinv, int n) {
  int i = blockIdx.x * blockDim.x + threadIdx.x;
  if (i < n) dinv[i] = rsqrtf((float)cnt[i] + 1.0f);  // +1 self-loop; deg >= 1 always
}

// single-workgroup prefix scan of cnt -> rowstart[0..n], rowstart[n] = E
__global__ __launch_bounds__(1024) void k_scan(const int* __restrict__ cnt,
                                               int* __restrict__ rs, int n) {
  __shared__ int part[1024];
  int t = threadIdx.x;
  int chunk = (n + 1023) >> 10;
  int lo = t * chunk;
  int hi = min(lo + chunk, n);
  int sum = 0;
  for (int i = lo; i < hi; ++i) sum += cnt[i];
  part[t] = sum;
  __syncthreads();
  for (int off = 1; off < 1024; off <<= 1) {
    int v = (t >= off) ? part[t - off] : 0;
    __syncthreads();
    part[t] += v;
    __syncthreads();
  }
  int base = (t > 0) ? part[t - 1] : 0;
  for (int i = lo; i < hi; ++i) { rs[i] = base; base += cnt[i]; }
  if (t == 1023) rs[n] = part[1023];
}

__global__ void k_fill(const int* __restrict__ src, const int* __restrict__ dst,
                       const float* __restrict__ dinv, const int* __restrict__ rs,
                       int* __restrict__ cur, int* __restrict__ csrc,
                       float* __restrict__ cw, int e) {
  int i = blockIdx.x * blockDim.x + threadIdx.x;
  if (i >= e) return;
  int d = dst[i], s = src[i];
  int p = atomicAdd(&cur[d], 1);
  int slot = rs[d] + p;
  csrc[slot] = s;
  cw[slot] = dinv[s] * dinv[d];
}

// transpose (and zero-pad) W [K x cin] row-major -> WT [cout x K] row-major
__global__ void k_prepWT(const float* __restrict__ W, float* __restrict__ WT,
                         int cin, int cout) {
  int i = blockIdx.x * blockDim.x + threadIdx.x;  // over cout*K, coalesced WT writes
  if (i >= cout * NF) return;
  int c = i / NF, k = i % NF;
  WT[i] = (c < cin) ? W[(size_t)k * cin + c] : 0.0f;
}

// ---------------- fp32 WMMA GEMM ----------------
// C[n x ncols] = A[n x 128] * W[128 x ncols], with W pre-transposed as WT[ncols x 128].
// One wave computes a 16 x (CT*16) strip: CT independent accumulator chains.
// Per k-step: 1+CT contiguous v2f fragment loads (one clause, immediate offsets),
// then CT WMMAs -> load latency overlaps matrix math.
template <int CT>
__global__ __launch_bounds__(256) void k_gemm_wmma(
    const float* __restrict__ A, const float* __restrict__ WT,
    float* __restrict__ C, int n, int ncols) {
  const int strips = ncols / (CT * 16);
  int wave = (int)((blockIdx.x * blockDim.x + threadIdx.x) >> 5);  // wave-uniform
  int lane = threadIdx.x & 31;
  int tilesM = n >> 4;
  int strip = wave % strips;
  int tM    = wave / strips;
  if (tM >= tilesM) return;  // wave-uniform exit: EXEC all-ones at every WMMA

  int half = lane >> 4;
  int l16  = lane & 15;
  int rowA    = (tM << 4) + l16;
  int colbase = strip * (CT * 16);

  v8f acc[CT];
#pragma unroll
  for (int t = 0; t < CT; ++t) acc[t] = (v8f){};

  // A fragment: lane holds K = {k0, k0+1}, k0 = kk + 2*half  (VGPR0=K{0,2}, VGPR1=K{1,3})
  // B fragment: identical form on transposed weights (contiguous pair per lane)
  const float* ap = A  + (size_t)rowA * NF + (half << 1);
  const float* bp = WT + (size_t)(colbase + l16) * NF + (half << 1);

#pragma unroll 4
  for (int kk = 0; kk < NF; kk += 4, ap += 4, bp += 4) {
    v2f a = *(const v2f*)ap;
    v2f b[CT];
#pragma unroll
    for (int t = 0; t < CT; ++t)
      b[t] = *(const v2f*)(bp + t * (16 * NF));  // +t*8192B immediate offset
#pragma unroll
    for (int t = 0; t < CT; ++t)
      acc[t] = __builtin_amdgcn_wmma_f32_16x16x4_f32(
          /*neg_a=*/false, a, /*neg_b=*/false, b[t],
          /*c_mod=*/(short)0, acc[t], /*reuse_a=*/false, /*reuse_b=*/false);
  }

  int rbase = (tM << 4) + (half << 3);  // C/D: VGPR r -> M = r + 8*half
#pragma unroll
  for (int t = 0; t < CT; ++t) {
    int cb = colbase + t * 16 + l16;
#pragma unroll
    for (int r = 0; r < 8; ++r)
      C[(size_t)(rbase + r) * ncols + cb] = acc[t][r];
  }
}

// ---------------- CSR gather-aggregation, 128-wide: one wave per node ----------------
__global__ __launch_bounds__(256) void k_agg128(
    const float* __restrict__ H, const int* __restrict__ rs,
    const int* __restrict__ csrc, const float* __restrict__ cw,
    const float* __restrict__ dinv, const float* __restrict__ bias,
    float* __restrict__ O, int n, int do_relu) {
  int node = (int)((blockIdx.x * blockDim.x + threadIdx.x) >> 5);
  int lane = threadIdx.x & 31;
  if (node >= n) return;
  int f = lane << 2;  // 4 contiguous features per lane -> coalesced float4
  float di = dinv[node];
  float sw = di * di;  // self-loop weight
  float4 hv = *(const float4*)(H + (size_t)node * NF + f);
  float ax = sw * hv.x, ay = sw * hv.y, az = sw * hv.z, aw = sw * hv.w;
  int s0 = rs[node], s1 = rs[node + 1];
  for (int s = s0; s < s1; ++s) {
    int sr  = csrc[s];
    float w = cw[s];
    float4 v = *(const float4*)(H + (size_t)sr * NF + f);
    ax += w * v.x; ay += w * v.y; az += w * v.z; aw += w * v.w;
  }
  float4 bv = *(const float4*)(bias + f);
  ax += bv.x; ay += bv.y; az += bv.z; aw += bv.w;
  if (do_relu) {
    ax = fmaxf(ax, 0.0f); ay = fmaxf(ay, 0.0f);
    az = fmaxf(az, 0.0f); aw = fmaxf(aw, 0.0f);
  }
  float4 o; o.x = ax; o.y = ay; o.z = az; o.w = aw;
  *(float4*)(O + (size_t)node * NF + f) = o;
}

// ---------------- final layer aggregation: 40 features, H stride = 48 ----------------
// Branchless inner loop: lanes 8..31 compute a discarded second accumulator.
__global__ __launch_bounds__(256) void k_agg40(
    const float* __restrict__ H, const int* __restrict__ rs,
    const int* __restrict__ csrc, const float* __restrict__ cw,
    const float* __restrict__ dinv, const float* __restrict__ bias,
    float* __restrict__ O, int n) {
  int node = (int)((blockIdx.x * blockDim.x + threadIdx.x) >> 5);
  int lane = threadIdx.x & 31;
  if (node >= n) return;
  int f0 = lane;
  int f1 = lane + 32;
  bool v1 = (f1 < NC);
  int f1c = v1 ? f1 : 0;  // clamped: loads always in-bounds, result discarded
  float di = dinv[node];
  float sw = di * di;
  const float* hp = H + (size_t)node * NCP;
  float a0 = sw * hp[f0];
  float a1 = sw * hp[f1c];
  int s0 = rs[node], s1 = rs[node + 1];
  for (int s = s0; s < s1; ++s) {
    int sr  = csrc[s];
    float w = cw[s];
    const float* q = H + (size_t)sr * NCP;
    a0 += w * q[f0];
    a1 += w * q[f1c];
  }
  O[(size_t)node * NC + f0] = a0 + bias[f0];
  if (v1) O[(size_t)node * NC + f1] = a1 + bias[f1];
}

// ---------------- host launcher ----------------

static inline size_t alignup(size_t x) { return (x + 255) & ~(size_t)255; }
static inline int cdiv(int a, int b) { return (a + b - 1) / b; }

extern "C" void kernel_launch(void* const* d_in, const int* in_sizes, int n_in,
                              void* d_out, int out_size, void* d_ws, size_t ws_size,
                              hipStream_t stream) {
  const float* x  = (const float*)d_in[0];
  const int*   ei = (const int*)d_in[1];
  const int* srcv = ei;
  const int* dstv = ei + NE;
  const float* W1 = (const float*)d_in[2]; const float* b1 = (const float*)d_in[3];
  const float* W2 = (const float*)d_in[4]; const float* b2 = (const float*)d_in[5];
  const float* W3 = (const float*)d_in[6]; const float* b3 = (const float*)d_in[7];
  const float* W4 = (const float*)d_in[8]; const float* b4 = (const float*)d_in[9];

  float* out_logits = (float*)d_out;                 // [NN x 40]
  float* x_latent   = out_logits + (size_t)NN * NC;  // [NN x 128]

  char* p = (char*)d_ws;
  int*   cnt  = (int*)p;    p += alignup((size_t)NN * 4);
  int*   cur  = (int*)p;    p += alignup((size_t)NN * 4);
  float* dinv = (float*)p;  p += alignup((size_t)NN * 4);
  int*   rs   = (int*)p;    p += alignup(((size_t)NN + 1) * 4);
  int*   csrc = (int*)p;    p += alignup((size_t)NE * 4);
  float* cw   = (float*)p;  p += alignup((size_t)NE * 4);
  float* WT1  = (float*)p;  p += alignup((size_t)NF * NF * 4);
  float* WT2  = (float*)p;  p += alignup((size_t)NF * NF * 4);
  float* WT3  = (float*)p;  p += alignup((size_t)NF * NF * 4);
  float* WT4  = (float*)p;  p += alignup((size_t)NCP * NF * 4);
  float* LIN  = (float*)p;  p += alignup((size_t)NN * NF * 4);
  float* AGG  = (float*)p;  p += alignup((size_t)NN * NF * 4);

  // --- CSR (by destination) + norms + transposed/padded weights, rebuilt every call ---
  k_init  <<<cdiv(NN, 256), 256, 0, stream>>>(cnt, cur, NN);
  k_count <<<cdiv(NE, 256), 256, 0, stream>>>(dstv, cnt, NE);
  k_dinv  <<<cdiv(NN, 256), 256, 0, stream>>>(cnt, dinv, NN);
  k_scan  <<<1, 1024, 0, stream>>>(cnt, rs, NN);
  k_fill  <<<cdiv(NE, 256), 256, 0, stream>>>(srcv, dstv, dinv, rs, cur, csrc, cw, NE);
  k_prepWT<<<cdiv(NF * NF, 256),  256, 0, stream>>>(W1, WT1, NF, NF);
  k_prepWT<<<cdiv(NF * NF, 256),  256, 0, stream>>>(W2, WT2, NF, NF);
  k_prepWT<<<cdiv(NF * NF, 256),  256, 0, stream>>>(W3, WT3, NF, NF);
  k_prepWT<<<cdiv(NCP * NF, 256), 256, 0, stream>>>(W4, WT4, NC, NCP);

  const int aggBlocks = cdiv(NN * 32, 256);  // one wave per node
  // layers 1-3: CT=4 -> 2 strips per 16-row tile
  const int gemmBlocksH = cdiv((NN / 16) * (NF / (4 * 16)), 8);
  // layer 4:   CT=3 -> 1 strip per 16-row tile (48 padded cols)
  const int gemmBlocksC = cdiv((NN / 16) * (NCP / (3 * 16)), 8);

  // layer 1
  k_gemm_wmma<4><<<gemmBlocksH, 256, 0, stream>>>(x,   WT1, LIN, NN, NF);
  k_agg128<<<aggBlocks, 256, 0, stream>>>(LIN, rs, csrc, cw, dinv, b1, AGG, NN, 1);
  // layer 2
  k_gemm_wmma<4><<<gemmBlocksH, 256, 0, stream>>>(AGG, WT2, LIN, NN, NF);
  k_agg128<<<aggBlocks, 256, 0, stream>>>(LIN, rs, csrc, cw, dinv, b2, AGG, NN, 1);
  // layer 3 -> x_latent lives directly in d_out
  k_gemm_wmma<4><<<gemmBlocksH, 256, 0, stream>>>(AGG, WT3, LIN, NN, NF);
  k_agg128<<<aggBlocks, 256, 0, stream>>>(LIN, rs, csrc, cw, dinv, b3, x_latent, NN, 1);
  // layer 4: padded 48-col GEMM, then 40-wide aggregation
  k_gemm_wmma<3><<<gemmBlocksC, 256, 0, stream>>>(x_latent, WT4, LIN, NN, NCP);
  k_agg40<<<aggBlocks, 256, 0, stream>>>(LIN, rs, csrc, cw, dinv, b4, out_logits, NN);

  (void)in_sizes; (void)n_in; (void)out_size; (void)ws_size;
}